// MatrixFactorization_76630806495462
// MI455X (gfx1250) — compile-verified
//
#include <hip/hip_runtime.h>

typedef __attribute__((ext_vector_type(2))) float v2f;
typedef __attribute__((ext_vector_type(8))) float v8f;

#define EMB_D 64
#define NGEN  18

// Pre-pass: row-sums of genre_emb table (18 rows x 64) -> d_ws[0..17]
__global__ __launch_bounds__(64)
void genre_rowsum_kernel(const float* __restrict__ genre_emb,
                         float* __restrict__ rowsum) {
    int j = threadIdx.x;
    if (j < NGEN) {
        float s = 0.f;
        #pragma unroll
        for (int d = 0; d < EMB_D; ++d) s += genre_emb[j * EMB_D + d];
        rowsum[j] = s;
    }
}

// One wave (32 lanes) handles 16 rows. Dot products via chained
// V_WMMA_F32_16X16X4_F32 computing diag(Q * (I+O)^T).
__global__ __launch_bounds__(256)
void mf_score_kernel(const int*   __restrict__ user_id,
                     const int*   __restrict__ item_id,
                     const int*   __restrict__ occupation,
                     const int*   __restrict__ genre,
                     const float* __restrict__ user_emb,
                     const float* __restrict__ user_b,
                     const float* __restrict__ item_emb,
                     const float* __restrict__ item_b,
                     const float* __restrict__ occ_emb,
                     const float* __restrict__ occ_b,
                     const float* __restrict__ genre_rowsum,
                     const float* __restrict__ genre_b,
                     float*       __restrict__ out,
                     int Btot) {
    const int lane  = threadIdx.x & 31;
    const int wave  = threadIdx.x >> 5;
    const int r0    = (blockIdx.x * 8 + wave) * 16;   // first row of this wave
    if (r0 >= Btot) return;                            // uniform per wave: EXEC stays all-1

    const int row16 = lane & 15;                       // M / N index
    const int half  = lane >> 4;                       // selects K pair within 4-chunk
    const int row   = r0 + row16;

    const int uid = user_id[row];
    const int iid = item_id[row];
    const int occ = occupation[row];

    const float* qp = user_emb + (long)uid * EMB_D + 2 * half;
    const float* ip = item_emb + (long)iid * EMB_D + 2 * half;
    const float* op = occ_emb  + (long)occ * EMB_D + 2 * half;

    v8f   acc  = {};
    float sumI = 0.f;

    // C[m][n] += sum_k Q[m][k] * S[n][k],  S = I + O.  A frag: Q, B frag: S^T.
    // Per-lane fragment = float2 at (row16*64 + k0 + 2*half): x -> vgpr0, y -> vgpr1.
    #pragma unroll
    for (int k0 = 0; k0 < EMB_D; k0 += 4) {
        v2f a  = *(const v2f*)(qp + k0);
        v2f iv = *(const v2f*)(ip + k0);
        v2f ov = *(const v2f*)(op + k0);
        v2f b  = iv + ov;
        sumI  += iv.x + iv.y;
        acc = __builtin_amdgcn_wmma_f32_16x16x4_f32(
            /*neg_a=*/false, a, /*neg_b=*/false, b,
            /*c_mod=*/(short)0, acc, /*reuse_a=*/false, /*reuse_b=*/false);
    }
    // Halves each covered half the K dims of I; partner lane (l^16) has same row.
    sumI += __shfl_xor(sumI, 16, 32);

    // Genre scalar means (branchless): g = gsum/(cnt*64), bg = gbsum/(cnt*64).
    // cnt >= 1 guaranteed by the reference setup.
    float gsum = 0.f, gbsum = 0.f, fcnt = 0.f;
    #pragma unroll
    for (int j = 0; j < NGEN; ++j) {
        const float m = (genre[row * NGEN + j] != 0) ? 1.f : 0.f;
        gsum  = __builtin_fmaf(m, genre_rowsum[j], gsum);
        gbsum = __builtin_fmaf(m, genre_b[j],      gbsum);
        fcnt += m;
    }
    const float inv   = 1.f / (fcnt * (float)EMB_D);
    const float extra = user_b[uid] + item_b[iid] + occ_b[occ]
                      + (gsum * sumI + gbsum) * inv;

    // Diagonal extraction: lane holds C[r + 8*half][row16] in acc[r];
    // diag for row m lives on lane m (m<8) or lane m+16 (m>=8).
    #pragma unroll
    for (int r = 0; r < 8; ++r) {
        if (row16 == r + 8 * half) {
            out[row] = acc[r] + extra;
        }
    }
}

extern "C" void kernel_launch(void* const* d_in, const int* in_sizes, int n_in,
                              void* d_out, int out_size, void* d_ws, size_t ws_size,
                              hipStream_t stream) {
    const int*   user_id    = (const int*)  d_in[0];
    const int*   item_id    = (const int*)  d_in[1];
    const int*   occupation = (const int*)  d_in[2];
    const int*   genre      = (const int*)  d_in[3];
    const float* user_emb   = (const float*)d_in[4];
    const float* user_b     = (const float*)d_in[5];
    const float* item_emb   = (const float*)d_in[6];
    const float* item_b     = (const float*)d_in[7];
    const float* occ_emb    = (const float*)d_in[8];
    const float* occ_b      = (const float*)d_in[9];
    const float* genre_emb  = (const float*)d_in[10];
    const float* genre_b    = (const float*)d_in[11];
    float*       out        = (float*)d_out;
    float*       rowsum     = (float*)d_ws;           // 18 floats of scratch

    const int B = in_sizes[0];

    genre_rowsum_kernel<<<1, 64, 0, stream>>>(genre_emb, rowsum);

    const int waves  = (B + 15) / 16;      // 16 rows per wave
    const int blocks = (waves + 7) / 8;    // 8 waves (256 threads) per block
    mf_score_kernel<<<blocks, 256, 0, stream>>>(
        user_id, item_id, occupation, genre,
        user_emb, user_b, item_emb, item_b,
        occ_emb, occ_b, rowsum, genre_b, out, B);
}